// GNNLocal_42460046688958
// MI455X (gfx1250) — compile-verified
//
#include <hip/hip_runtime.h>
#include <hip/hip_bf16.h>
#include <math.h>

typedef __attribute__((ext_vector_type(2))) float v2f;
typedef __attribute__((ext_vector_type(8))) float v8f;
typedef int v4i_b128 __attribute__((vector_size(16)));   // matches builtin pointee

#if __has_builtin(__builtin_amdgcn_global_load_async_to_lds_b128) && \
    __has_builtin(__builtin_amdgcn_s_wait_asynccnt)
#define USE_ASYNC_LDS 1
#else
#define USE_ASYNC_LDS 0
#endif

#if USE_ASYNC_LDS
__device__ __forceinline__ void async_copy16(const float* g, float* l) {
    auto gp = (__attribute__((address_space(1))) v4i_b128*)g;
    auto lp = (__attribute__((address_space(3))) v4i_b128*)l;
    __builtin_amdgcn_global_load_async_to_lds_b128(gp, lp, 0, 0);
}
#endif

// ---------------------------------------------------------------------------
// Edge-weight max reduction (edge_attr is int in [0,10))
// ---------------------------------------------------------------------------
__global__ void ea_max_kernel(const int* __restrict__ ea, int* __restrict__ maxp, int E) {
    int tid    = blockIdx.x * blockDim.x + threadIdx.x;
    int stride = gridDim.x * blockDim.x;
    int m = 0;
    for (int i = tid; i < E; i += stride) m = max(m, ea[i]);
    for (int off = 16; off > 0; off >>= 1) m = max(m, __shfl_xor(m, off, 32));
    if ((threadIdx.x & 31) == 0) atomicMax(maxp, m);
}

// deg[col[e]] += edge_attr[e] / max   (f32 atomic, L2-resident)
__global__ void deg_kernel(const int* __restrict__ col, const int* __restrict__ ea,
                           const int* __restrict__ maxp, float* deg, int E) {
    int e = blockIdx.x * blockDim.x + threadIdx.x;
    if (e >= E) return;
    float inv = 1.0f / (float)(*maxp);
    float ew  = (float)ea[e] * inv;
    __hip_atomic_fetch_add(&deg[col[e]], ew, __ATOMIC_RELAXED, __HIP_MEMORY_SCOPE_AGENT);
}

// dis = deg > 0 ? rsqrt(deg) : 0   (in place)
__global__ void dis_kernel(float* deg_dis, int n) {
    int i = blockIdx.x * blockDim.x + threadIdx.x;
    if (i >= n) return;
    float d = deg_dis[i];
    deg_dis[i] = (d > 0.0f) ? rsqrtf(d) : 0.0f;
}

// norm[e] = dis[row]*ew*dis[col]
__global__ void norm_kernel(const int* __restrict__ row, const int* __restrict__ col,
                            const int* __restrict__ ea, const int* __restrict__ maxp,
                            const float* __restrict__ dis, float* __restrict__ nrm, int E) {
    int e = blockIdx.x * blockDim.x + threadIdx.x;
    if (e >= E) return;
    float inv = 1.0f / (float)(*maxp);
    float ew  = (float)ea[e] * inv;
    nrm[e] = dis[row[e]] * ew * dis[col[e]];
}

// h0[i][:] = emb[x[i]][:]   one thread per 4 floats
__global__ void gather_kernel(const int* __restrict__ x, const float* __restrict__ emb,
                              float* __restrict__ h0, int n) {
    int t = blockIdx.x * blockDim.x + threadIdx.x;
    int node = t >> 5;
    int part = t & 31;
    if (node >= n) return;
    const float4 v = *(const float4*)(emb + (unsigned)x[node] * 128u + part * 4);
    *(float4*)(h0 + (unsigned)node * 128u + part * 4) = v;
}

// ---------------------------------------------------------------------------
// Edge scatter: dst[col[e]][:] += norm[e] * src[row[e]][:]
// One wave per edge: coalesced float4 row read + 4 native f32 atomics/lane.
// ---------------------------------------------------------------------------
__global__ __launch_bounds__(256)
void scatter_kernel(const float* __restrict__ src, const int* __restrict__ row,
                    const int* __restrict__ col, const float* __restrict__ nrm,
                    float* dst, int E) {
    int gtid = blockIdx.x * blockDim.x + threadIdx.x;
    int e    = gtid >> 5;
    int lane = threadIdx.x & 31;
    if (e >= E) return;
    int   r = row[e];
    int   c = col[e];
    float w = nrm[e];
    const float4 v = *(const float4*)(src + (unsigned)r * 128u + lane * 4);
    float* d = dst + (unsigned)c * 128u + lane * 4;
    __hip_atomic_fetch_add(d + 0, v.x * w, __ATOMIC_RELAXED, __HIP_MEMORY_SCOPE_AGENT);
    __hip_atomic_fetch_add(d + 1, v.y * w, __ATOMIC_RELAXED, __HIP_MEMORY_SCOPE_AGENT);
    __hip_atomic_fetch_add(d + 2, v.z * w, __ATOMIC_RELAXED, __HIP_MEMORY_SCOPE_AGENT);
    __hip_atomic_fetch_add(d + 3, v.w * w, __ATOMIC_RELAXED, __HIP_MEMORY_SCOPE_AGENT);
}

// ---------------------------------------------------------------------------
// Fused WMMA GEMM: out[N x Kout] = ACT( A[N x K] @ W[K x Kout] (+ addv) (+ bias) )
// Block = one 16-row M tile x full Kout; one wave per 16-col N tile.
// A tile AND full W panel staged in LDS via async b128 copies (ASYNCcnt path),
// inner loop is pure ds_load + v_wmma_f32_16x16x4_f32 (full fp32 numerics).
// LDS strides: A = K+4 words (16B rows, conflict-free A frags),
//              W = Kout+8 words (16B rows, lane-halves in disjoint bank ranges).
// ACT: 0 = none, 1 = relu, 2 = tanh
// ---------------------------------------------------------------------------
template <int K, int Kout, int ACT>
__global__ __launch_bounds__(256)
void gemm_kernel(const float* __restrict__ A, const float* __restrict__ W,
                 const float* addv, const float* __restrict__ bias, float* out) {
    constexpr int SA   = K + 4;          // A tile stride (words)
    constexpr int SW   = Kout + 8;       // W stride (words)
    constexpr int BDIM = 2 * Kout;       // 32 * (Kout/16) threads
    extern __shared__ float lds[];
    float* ldsA = lds;
    float* ldsW = lds + 16 * SA;

    const int m0  = blockIdx.x * 16;
    const int tid = threadIdx.x;

#if USE_ASYNC_LDS
    // Stage A tile (16 x K) as 16B chunks
    constexpr int ACH = K / 4;           // chunks per A row
    for (int i = tid; i < 16 * ACH; i += BDIM) {
        int r = i / ACH, c = i % ACH;    // compile-time -> shifts
        async_copy16(A + (unsigned)(m0 + r) * K + 4 * c, ldsA + r * SA + 4 * c);
    }
    // Stage W panel (K x Kout)
    constexpr int WCH = Kout / 4;
    for (int i = tid; i < K * WCH; i += BDIM) {
        int r = i / WCH, c = i % WCH;
        async_copy16(W + (unsigned)r * Kout + 4 * c, ldsW + r * SW + 4 * c);
    }
    __builtin_amdgcn_s_wait_asynccnt(0);
    __syncthreads();
#else
    for (int i = tid; i < 16 * K; i += BDIM) {
        int r = i / K, c = i % K;
        ldsA[r * SA + c] = A[(unsigned)(m0 + r) * K + c];
    }
    for (int i = tid; i < K * Kout; i += BDIM) {
        int r = i / Kout, c = i % Kout;
        ldsW[r * SW + c] = W[i];
    }
    __syncthreads();
#endif

    const int wave = tid >> 5;
    const int lane = tid & 31;
    const int half = lane >> 4;          // 0: lanes 0-15, 1: lanes 16-31
    const int lr   = lane & 15;
    const int n0   = wave << 4;

    v8f acc = {};
    #pragma unroll
    for (int kk = 0; kk < K; kk += 4) {
        const int ka = kk + 2 * half;    // half 0 -> K=kk,kk+1; half 1 -> kk+2,kk+3
        v2f a, b;
        a.x = ldsA[lr * SA + ka];
        a.y = ldsA[lr * SA + ka + 1];
        b.x = ldsW[ka * SW + n0 + lr];
        b.y = ldsW[(ka + 1) * SW + n0 + lr];
        acc = __builtin_amdgcn_wmma_f32_16x16x4_f32(
            /*neg_a=*/false, a, /*neg_b=*/false, b,
            /*c_mod=*/(short)0, acc, /*reuse_a=*/false, /*reuse_b=*/false);
    }

    const float bv = bias ? bias[n0 + lr] : 0.0f;
    #pragma unroll
    for (int v = 0; v < 8; ++v) {
        const int      m   = m0 + v + 8 * half;   // lanes 0-15: M=v; 16-31: M=8+v
        const unsigned off = (unsigned)m * Kout + n0 + lr;
        float val = acc[v] + bv;
        if (addv) val += addv[off];
        if (ACT == 1)      val = fmaxf(val, 0.0f);
        else if (ACT == 2) val = tanhf(val);
        out[off] = val;
    }
}

// ---------------------------------------------------------------------------
// Host-side orchestration
// ---------------------------------------------------------------------------
extern "C" void kernel_launch(void* const* d_in, const int* in_sizes, int n_in,
                              void* d_out, int out_size, void* d_ws, size_t ws_size,
                              hipStream_t stream) {
    const int N   = in_sizes[0];          // 100000 nodes
    const int E   = in_sizes[2];          // 625000 edges
    const int HID = 128;

    const int*   x       = (const int*)d_in[0];
    const int*   erow    = (const int*)d_in[1];          // edge_index[0]
    const int*   ecol    = ((const int*)d_in[1]) + E;    // edge_index[1]
    const int*   ea      = (const int*)d_in[2];
    const float* emb     = (const float*)d_in[3];
    const float* init_w  = (const float*)d_in[4];
    const float* ws_l    = (const float*)d_in[5];        // [3,128,128]
    const float* root_ws = (const float*)d_in[6];        // [4,128,128]
    const float* biases  = (const float*)d_in[7];        // [4,128]
    const float* lin1_w  = (const float*)d_in[8];
    const float* lin1_b  = (const float*)d_in[9];
    const float* lin2_w  = (const float*)d_in[10];
    const float* lin2_b  = (const float*)d_in[11];
    const float* lin3_w  = (const float*)d_in[12];
    const float* lin3_b  = (const float*)d_in[13];
    float*       outp    = (float*)d_out;

    // Scratch carve-up (~157 MB; whole working set is L2-resident on 192MB L2)
    char*  p    = (char*)d_ws;
    float* h0   = (float*)p;  p += (size_t)N * HID * 4;
    float* cur  = (float*)p;  p += (size_t)N * HID * 4;  // also the scatter accumulator
    float* tmp  = (float*)p;  p += (size_t)N * HID * 4;
    float* dis  = (float*)p;  p += (size_t)N * 4;
    float* nrm  = (float*)p;  p += (size_t)E * 4;
    int*   maxp = (int*)p;

    const size_t nodeBytes = (size_t)N * HID * 4;

    // --- edge normalization --------------------------------------------------
    (void)hipMemsetAsync(maxp, 0, sizeof(int), stream);
    ea_max_kernel<<<256, 256, 0, stream>>>(ea, maxp, E);
    (void)hipMemsetAsync(dis, 0, (size_t)N * 4, stream);
    deg_kernel<<<(E + 255) / 256, 256, 0, stream>>>(ecol, ea, maxp, dis, E);
    dis_kernel<<<(N + 255) / 256, 256, 0, stream>>>(dis, N);
    norm_kernel<<<(E + 255) / 256, 256, 0, stream>>>(erow, ecol, ea, maxp, dis, nrm, E);

    // --- embedding gather ----------------------------------------------------
    gather_kernel<<<((size_t)N * 32 + 255) / 256, 256, 0, stream>>>(x, emb, h0, N);

    const int mTiles = N / 16;            // 6250
    const int lds_128_128 = (16 * (128 + 4) + 128 * (128 + 8)) * 4;  // 78080 B
    const int lds_128_64  = (16 * (128 + 4) + 128 * (64 + 8)) * 4;   // 45312 B
    const int lds_64_32   = (16 * (64 + 4)  + 64 * (32 + 8)) * 4;    // 14592 B
    const int scatBlocks  = (int)(((size_t)E * 32 + 255) / 256);

    // --- ARMA layers ---------------------------------------------------------
    for (int t = 0; t < 4; ++t) {
        const float* Win = (t == 0) ? init_w : (ws_l + (size_t)(t - 1) * HID * HID);
        const float* Ain = (t == 0) ? h0 : cur;
        // tmp = Ain @ Win
        gemm_kernel<128, 128, 0><<<mTiles, 256, lds_128_128, stream>>>(
            Ain, Win, nullptr, nullptr, tmp);
        // cur = scatter-add of norm-weighted tmp rows
        (void)hipMemsetAsync(cur, 0, nodeBytes, stream);
        scatter_kernel<<<scatBlocks, 256, 0, stream>>>(tmp, erow, ecol, nrm, cur, E);
        // cur = relu(cur + h0 @ root_ws[t] + biases[t])   (in-place add)
        gemm_kernel<128, 128, 1><<<mTiles, 256, lds_128_128, stream>>>(
            h0, root_ws + (size_t)t * HID * HID, cur, biases + (size_t)t * HID, cur);
    }

    // --- MLP head ------------------------------------------------------------
    // tmp = relu(cur @ lin1_w + lin1_b)          [N,128]
    gemm_kernel<128, 128, 1><<<mTiles, 256, lds_128_128, stream>>>(
        cur, lin1_w, nullptr, lin1_b, tmp);
    // cur = relu(tmp @ lin2_w + lin2_b)          [N,64]
    gemm_kernel<128, 64, 1><<<mTiles, 128, lds_128_64, stream>>>(
        tmp, lin2_w, nullptr, lin2_b, cur);
    // out = tanh(cur @ lin3_w + lin3_b)          [N,32]
    gemm_kernel<64, 32, 2><<<mTiles, 64, lds_64_32, stream>>>(
        cur, lin3_w, nullptr, lin3_b, outp);
}